// TemporalPooling_spk_83124797047406
// MI455X (gfx1250) — compile-verified
//
#include <hip/hip_runtime.h>
#include <hip/hip_bf16.h>

// ---------------- problem constants ----------------
#define BATCH   8
#define CDIM    1024
#define TDIM    2048            // T after 4:1 pooling
#define SEG     (BATCH * CDIM * TDIM)   // 16,777,216 floats per output tensor
#define MAXLEN  16

// ---------------- GEMM tiling ----------------
#define BM   128                // workgroup M tile (output channel d)
#define BN   64                 // workgroup N tile (time t)
#define KC   32                 // K chunk
#define SA   (KC + 2)           // LDS row stride for A (34 = TDM pad 2 dwords / 32)
#define SB   (BN + 16)          // LDS row stride for B (80 = TDM pad 16 dwords / 64)
#define NK   (CDIM / KC)        // 32 K chunks

typedef __attribute__((ext_vector_type(2))) float v2f;
typedef __attribute__((ext_vector_type(8))) float v8f;
typedef unsigned int v4u __attribute__((ext_vector_type(4)));
typedef unsigned int v8u __attribute__((ext_vector_type(8)));

// ============================================================
// Kernel 1: 4:1 average pool.  x (B,C,8192) -> tap (B,C,2048)
// ============================================================
__global__ void tp_pool4(const float4* __restrict__ x4, float* __restrict__ tap) {
    int i = blockIdx.x * 256 + threadIdx.x;     // exactly SEG threads
    float4 v = x4[i];
    tap[i] = (v.x + v.y + v.z + v.w) * 0.25f;
}

// ------------------------------------------------------------
// TDM 2D tile load via inline asm.  The destination LDS pointer is
// passed as an (unreferenced) operand so the shared array's address
// escapes into the asm: combined with the "memory" clobber this makes
// the compiler treat the asm as a writer of the LDS array, keeping
// the ds_load fragment reads alive (without it, a never-stored LDS
// global lets GVN fold all LDS loads to undef).
// Descriptor bitfields per CDNA5 ISA ch.10.8 (D# groups 0/1);
// VADDR2/VADDR3 omitted (NULL) for 2D tensors.
// ------------------------------------------------------------
__device__ __forceinline__ void tdm_load_2d(const float* lds_dst, unsigned lds_byte_off,
                                            const void* gptr,
                                            unsigned tile_d0, unsigned tile_d1,
                                            unsigned long long stride0,
                                            unsigned pad_interval_code,
                                            unsigned pad_amount_code) {
    unsigned long long ga = (unsigned long long)(size_t)gptr;
    v4u g0;
    g0[0] = 1u;                                        // count=1, user descriptor
    g0[1] = lds_byte_off;                              // lds_addr
    g0[2] = (unsigned)(ga & 0xFFFFFFFFu);              // global_addr[31:0]
    g0[3] = (unsigned)((ga >> 32) & 0x01FFFFFFu)       // global_addr[56:32]
          | (2u << 30);                                // type = 2 ("image")
    v8u g1;
    g1[0] = (2u << 16)                                 // data_size = 4 bytes
          | (1u << 20)                                 // pad_enable
          | (pad_interval_code << 22)
          | (pad_amount_code << 25);
    g1[1] = (tile_d0 & 0xFFFFu) << 16;                 // tensor_dim0[15:0]
    g1[2] = (tile_d0 >> 16) | ((tile_d1 & 0xFFFFu) << 16); // dim0 hi | dim1 lo
    g1[3] = (tile_d1 >> 16) | (tile_d0 << 16);         // dim1 hi | tile_dim0
    g1[4] = tile_d1 & 0xFFFFu;                         // tile_dim1 (tile_dim2 = 0)
    g1[5] = (unsigned)(stride0 & 0xFFFFFFFFull);       // tensor_dim0_stride lo
    g1[6] = (unsigned)((stride0 >> 32) & 0xFFFFull);   // stride hi (dim1 stride unused)
    g1[7] = 0u;
    asm volatile("tensor_load_to_lds %0, %1"
                 :
                 : "s"(g0), "s"(g1), "v"(lds_dst)      // %2 unused: escapes LDS addr
                 : "memory");
}

__device__ __forceinline__ void tdm_wait0() {
    asm volatile("s_wait_tensorcnt 0x0" ::: "memory");
}

// ============================================================
// Kernel 2/3: Y(b) = W (CxC) * X(b) (CxT) + bias, fp32 WMMA.
// grid = (T/BN, C/BM, B), block = 256 (8 waves, wave32).
// Tile 0 staged by the waves; tiles 1..NK-1 DMA'd by the TDM
// into the other half of a double-buffered LDS while compute runs.
// ============================================================
__global__ __launch_bounds__(256)
void tp_gemm_bias(const float* __restrict__ W,
                  const float* __restrict__ bias,
                  const float* __restrict__ X,
                  float* __restrict__ Y) {
    // single LDS symbol -> known byte offsets for TDM descriptors
    __shared__ float lds[2 * BM * SA + 2 * KC * SB];
    constexpr unsigned ASZ = BM * SA;   // floats per A buffer
    constexpr unsigned BSZ = KC * SB;   // floats per B buffer

    const int tid  = threadIdx.x;
    const int lane = tid & 31;                  // wave32
    const int half = lane >> 4;
    const int lm   = lane & 15;
    const int kb   = half * 2;                  // K sub-offset in A/B fragment layout
    const int wave = tid >> 5;
    const int waveM = (wave & 3) * 32;          // 4 waves along M
    const int waveN = (wave >> 2) * 32;         // 2 waves along N

    const int t0 = blockIdx.x * BN;
    const int m0 = blockIdx.y * BM;
    const int b  = blockIdx.z;
    const size_t xbase = (size_t)b * CDIM * TDIM;

    const bool issuer = (wave == 0);            // one wave issues TDM ops
    auto issue_tiles = [&](int s, int buf) {
        const float* gA = W + (size_t)m0 * CDIM + s * KC;                // 128 x 32 tile
        const float* gB = X + xbase + (size_t)(s * KC) * TDIM + t0;      // 32 x 64 tile
        // A: 32 dwords/row, +2 dwords pad  -> LDS stride 34  (codes: 4 -> 32, 1 -> 2)
        tdm_load_2d(&lds[(unsigned)buf * ASZ], (unsigned)(buf * ASZ) * 4u,
                    gA, KC, BM, (unsigned long long)CDIM, 4u, 1u);
        // B: 64 dwords/row, +16 dwords pad -> LDS stride 80  (codes: 5 -> 64, 15 -> 16)
        tdm_load_2d(&lds[2 * ASZ + (unsigned)buf * BSZ],
                    (unsigned)(2 * ASZ + buf * BSZ) * 4u,
                    gB, BN, KC, (unsigned long long)TDIM, 5u, 15u);
    };

    // ---- prologue: tile 0 staged by threads (real stores to the LDS
    // array also pin it as "written" for alias analysis) ----
    {
        const int arow = tid >> 3, acol = (tid & 7) * 4;
        const int brow = tid >> 4, bcol = (tid & 15) * 4;
#pragma unroll
        for (int i = 0; i < 4; ++i) {
            float4 ra = *(const float4*)(W + (size_t)(m0 + arow + 32 * i) * CDIM + acol);
            float* p = &lds[(arow + 32 * i) * SA + acol];
            p[0] = ra.x; p[1] = ra.y; p[2] = ra.z; p[3] = ra.w;
        }
#pragma unroll
        for (int i = 0; i < 2; ++i) {
            float4 rb = *(const float4*)(X + xbase + (size_t)(brow + 16 * i) * TDIM + t0 + bcol);
            float* p = &lds[2 * ASZ + (brow + 16 * i) * SB + bcol];
            p[0] = rb.x; p[1] = rb.y; p[2] = rb.z; p[3] = rb.w;
        }
    }
    __syncthreads();

    v8f acc[2][2];
#pragma unroll
    for (int mi = 0; mi < 2; ++mi)
#pragma unroll
        for (int ni = 0; ni < 2; ++ni)
            acc[mi][ni] = (v8f)0.0f;

    for (int s = 0; s < NK; ++s) {
        const int buf = s & 1;
        if (s + 1 < NK && issuer) issue_tiles(s + 1, buf ^ 1);  // DMA overlaps compute

        const float* __restrict__ Ab = &lds[(unsigned)buf * ASZ];
        const float* __restrict__ Bb = &lds[2 * ASZ + (unsigned)buf * BSZ];
#pragma unroll
        for (int k = 0; k < KC; k += 4) {
            v2f af[2], bf[2];
#pragma unroll
            for (int mi = 0; mi < 2; ++mi) {
                const float* p = &Ab[(waveM + mi * 16 + lm) * SA + k + kb];
                af[mi].x = p[0];
                af[mi].y = p[1];
            }
#pragma unroll
            for (int ni = 0; ni < 2; ++ni) {
                const float* p = &Bb[(k + kb) * SB + waveN + ni * 16 + lm];
                bf[ni].x = p[0];
                bf[ni].y = p[SB];
            }
#pragma unroll
            for (int mi = 0; mi < 2; ++mi)
#pragma unroll
                for (int ni = 0; ni < 2; ++ni)
                    acc[mi][ni] = __builtin_amdgcn_wmma_f32_16x16x4_f32(
                        false, af[mi], false, bf[ni],
                        (short)0, acc[mi][ni], false, false);
        }
        __syncthreads();                 // everyone done reading buf

        if (s + 1 < NK) {
            if (issuer) tdm_wait0();     // next buffer landed
            __syncthreads();
        }
    }

    // ---- epilogue: D layout = VGPR r -> rows (r, r+8) across lane halves ----
#pragma unroll
    for (int mi = 0; mi < 2; ++mi) {
#pragma unroll
        for (int ni = 0; ni < 2; ++ni) {
            const int tcol = t0 + waveN + ni * 16 + lm;
#pragma unroll
            for (int r = 0; r < 8; ++r) {
                const int drow = m0 + waveM + mi * 16 + r + half * 8;
                Y[xbase + (size_t)drow * TDIM + tcol] = acc[mi][ni][r] + bias[drow];
            }
        }
    }
}

// ============================================================
// Kernel 4: trailing-window mean (len<=16), sample every 4th t,
// replicate x4.  One thread per group of 4 output times.
// ============================================================
__global__ void tp_window_gather(const float* __restrict__ ff2, float* __restrict__ ret) {
    int tid = blockIdx.x * 256 + threadIdx.x;   // B*C*(T/4) threads
    int g   = tid & (TDIM / 4 - 1);
    int row = tid >> 9;                          // T/4 = 512
    int s   = g * 4;
    int L   = (s + 1 < MAXLEN) ? (s + 1) : MAXLEN;
    const float* p = ff2 + (size_t)row * TDIM + s;
    float sum = 0.0f;
    for (int j = 0; j < L; ++j) sum += p[-j];
    float avg = sum / (float)L;
    float4 o = make_float4(avg, avg, avg, avg);
    *(float4*)(ret + (size_t)row * TDIM + s) = o;
}

// ============================================================
extern "C" void kernel_launch(void* const* d_in, const int* in_sizes, int n_in,
                              void* d_out, int out_size, void* d_ws, size_t ws_size,
                              hipStream_t stream) {
    const float* x  = (const float*)d_in[0];
    const float* W1 = (const float*)d_in[1];
    const float* b1 = (const float*)d_in[2];
    const float* W2 = (const float*)d_in[3];
    const float* b2 = (const float*)d_in[4];

    float* ret = (float*)d_out;          // output order: ret, tap, ff1, ff2
    float* tap = ret + (size_t)SEG;
    float* ff1 = tap + (size_t)SEG;
    float* ff2 = ff1 + (size_t)SEG;

    // 1) pool
    tp_pool4<<<SEG / 256, 256, 0, stream>>>((const float4*)x, tap);

    // 2) ff1 = W1*tap + b1 ; 3) ff2 = W2*ff1 + b2
    dim3 gg(TDIM / BN, CDIM / BM, BATCH);
    tp_gemm_bias<<<gg, 256, 0, stream>>>(W1, b1, tap, ff1);
    tp_gemm_bias<<<gg, 256, 0, stream>>>(W2, b2, ff1, ff2);

    // 4) windowed average + stride-4 gather + replicate
    tp_window_gather<<<(BATCH * CDIM * (TDIM / 4)) / 256, 256, 0, stream>>>(ff2, ret);
}